// PreFeatureExtractor_24352464569905
// MI455X (gfx1250) — compile-verified
//
#include <hip/hip_runtime.h>
#include <hip/hip_bf16.h>

#define B_   8
#define N_   4096
#define K_   20
#define OUTC 64

typedef float v2f __attribute__((ext_vector_type(2)));
typedef float v8f __attribute__((ext_vector_type(8)));

// Branchless sorted-descending insert: 1 cmp + 4 cndmask per slot, no phis.
__device__ __forceinline__ void topk_insert(float (&bd)[K_], int (&bi)[K_],
                                            float s, int idx)
{
    float cs = s; int ci = idx;
#pragma unroll
    for (int j = 0; j < K_; ++j) {
        const bool sw = cs > bd[j];
        const float nd = sw ? cs : bd[j];
        const float od = sw ? bd[j] : cs;
        const int   ni = sw ? ci : bi[j];
        const int   oi = sw ? bi[j] : ci;
        bd[j] = nd; bi[j] = ni;
        cs = od; ci = oi;
    }
}

// ---------------------------------------------------------------------------
// Kernel A: exact KNN via V_WMMA_F32_16X16X4_F32 Gram tiles.
// score[m,n] = cand_m . query_n - 0.5*|cand_m|^2  (maximize == nearest);
// the -0.5*|c|^2 term rides in A's K=3 slot (pre-scaled in LDS), B's K=3
// row is 1.0. Two tiles (32 candidates) per iteration: two independent
// WMMAs overlap the selection VALU of the other tile.
// ---------------------------------------------------------------------------
__global__ __launch_bounds__(256)
void knn_kernel(const float* __restrict__ x, int* __restrict__ nbr)
{
    __shared__ float4 pts[N_];   // x, y, z, -0.5*|p|^2  (64 KB)

    const int tid   = threadIdx.x;
    const int batch = blockIdx.x >> 5;           // 32 blocks per batch
    const int qblk  = (blockIdx.x & 31) * 128;   // 128 queries per block
    const int gbase = batch * N_;

    for (int i = tid; i < N_; i += 256) {
        const float px = x[(gbase + i) * 3 + 0];
        const float py = x[(gbase + i) * 3 + 1];
        const float pz = x[(gbase + i) * 3 + 2];
        pts[i] = make_float4(px, py, pz, -0.5f * (px * px + py * py + pz * pz));
    }
    __syncthreads();

    const int  wave = tid >> 5;
    const int  lane = tid & 31;
    const int  lo16 = lane & 15;
    const bool hi   = lane >= 16;
    const int  qloc = qblk + wave * 16 + lo16;   // this lane's query (batch-local)

    // B operand (4x16): lanes 0-15 -> (x,y); lanes 16-31 -> (z, 1)
    const float4 q = pts[qloc];
    v2f bop;
    bop.x = hi ? q.z : q.x;
    bop.y = hi ? 1.0f : q.y;

    float bd[K_];
    int   bi[K_];
#pragma unroll
    for (int j = 0; j < K_; ++j) { bd[j] = -3.4e38f; bi[j] = 0; }

    const int rowOff = hi ? 8 : 0;

    // software-pipelined, two Gram tiles per iteration
    float4 c0 = pts[lo16];
    float4 c1 = pts[16 + lo16];
    for (int t = 0; t < N_; t += 32) {
        const int tn = (t + 32) & (N_ - 1);      // wrap: branch-free prefetch
        const float4 n0 = pts[tn + lo16];
        const float4 n1 = pts[tn + 16 + lo16];

        // A operands (16x4): lanes 0-15 -> (x,y); lanes 16-31 -> (z, -0.5|c|^2)
        v2f a0, a1;
        a0.x = hi ? c0.z : c0.x;  a0.y = hi ? c0.w : c0.y;
        a1.x = hi ? c1.z : c1.x;  a1.y = hi ? c1.w : c1.y;

        v8f z0 = {0.f, 0.f, 0.f, 0.f, 0.f, 0.f, 0.f, 0.f};
        v8f z1 = {0.f, 0.f, 0.f, 0.f, 0.f, 0.f, 0.f, 0.f};
        const v8f acc0 = __builtin_amdgcn_wmma_f32_16x16x4_f32(
            false, a0, false, bop, (short)0, z0, false, false);
        const v8f acc1 = __builtin_amdgcn_wmma_f32_16x16x4_f32(
            false, a1, false, bop, (short)0, z1, false, false);

#pragma unroll
        for (int v = 0; v < 8; ++v) {
            const int cand = t + rowOff + v;
            float s = acc0[v];
            s = (cand == qloc) ? -3.4e38f : s;   // exclude self, branchless
            if (s > bd[K_ - 1])
                topk_insert(bd, bi, s, cand);
        }
#pragma unroll
        for (int v = 0; v < 8; ++v) {
            const int cand = t + 16 + rowOff + v;
            float s = acc1[v];
            s = (cand == qloc) ? -3.4e38f : s;
            if (s > bd[K_ - 1])
                topk_insert(bd, bi, s, cand);
        }
        c0 = n0; c1 = n1;
    }

    // merge the two half-lists: lanes 0-15 absorb lanes 16-31
#pragma unroll
    for (int j = 0; j < K_; ++j) {
        const float od = __shfl_xor(bd[j], 16, 32);
        const int   oi = __shfl_xor(bi[j], 16, 32);
        if (!hi && od > bd[K_ - 1])
            topk_insert(bd, bi, od, oi);
    }

    if (!hi) {
        int* o = nbr + (gbase + qloc) * K_;
#pragma unroll
        for (int j = 0; j < K_; ++j) o[j] = bi[j];
    }
}

// ---------------------------------------------------------------------------
// Kernel B: EdgeConv MLP + max:
//   out[o] = relu( b[o] + x_i.(u_o - v_o) + max_k( x_jk . v_o ) )
// Batch coords in LDS for the random neighbor gather; one point per thread.
// ---------------------------------------------------------------------------
__global__ __launch_bounds__(256)
void edge_mlp_kernel(const float* __restrict__ x, const float* __restrict__ W,
                     const float* __restrict__ bias, const int* __restrict__ nbr,
                     float* __restrict__ out)
{
    __shared__ float sx[N_], sy[N_], sz[N_];   // 48 KB
    __shared__ float aW[3][OUTC];              // u - v
    __shared__ float vW[3][OUTC];              // v
    __shared__ float bb[OUTC];

    const int tid   = threadIdx.x;
    const int batch = blockIdx.x >> 4;          // 16 blocks per batch
    const int pblk  = (blockIdx.x & 15) * 256;
    const int gbase = batch * N_;

    for (int i = tid; i < N_; i += 256) {
        sx[i] = x[(gbase + i) * 3 + 0];
        sy[i] = x[(gbase + i) * 3 + 1];
        sz[i] = x[(gbase + i) * 3 + 2];
    }
    if (tid < OUTC) {
        const float u0 = W[0 * OUTC + tid], u1 = W[1 * OUTC + tid], u2 = W[2 * OUTC + tid];
        const float v0 = W[3 * OUTC + tid], v1 = W[4 * OUTC + tid], v2 = W[5 * OUTC + tid];
        aW[0][tid] = u0 - v0; aW[1][tid] = u1 - v1; aW[2][tid] = u2 - v2;
        vW[0][tid] = v0;      vW[1][tid] = v1;      vW[2][tid] = v2;
        bb[tid] = bias[tid];
    }
    __syncthreads();

    const int   ploc = pblk + tid;
    const float xi0 = sx[ploc], xi1 = sy[ploc], xi2 = sz[ploc];

    float nx[K_], ny[K_], nz[K_];
    const int* nn = nbr + (gbase + ploc) * K_;
#pragma unroll
    for (int k = 0; k < K_; ++k) {
        const int j = nn[k];
        nx[k] = sx[j]; ny[k] = sy[j]; nz[k] = sz[j];
    }

    float* op = out + (gbase + ploc) * OUTC;
#pragma unroll 2
    for (int o = 0; o < OUTC; ++o) {
        const float a0 = aW[0][o], a1 = aW[1][o], a2 = aW[2][o];
        const float v0 = vW[0][o], v1 = vW[1][o], v2 = vW[2][o];
        const float base = bb[o] + xi0 * a0 + xi1 * a1 + xi2 * a2;
        float m0 = -3.4e38f, m1 = -3.4e38f;
#pragma unroll
        for (int k = 0; k < K_; k += 2) {
            m0 = fmaxf(m0, nx[k]     * v0 + ny[k]     * v1 + nz[k]     * v2);
            m1 = fmaxf(m1, nx[k + 1] * v0 + ny[k + 1] * v1 + nz[k + 1] * v2);
        }
        op[o] = fmaxf(base + fmaxf(m0, m1), 0.0f);
    }
}

// ---------------------------------------------------------------------------
extern "C" void kernel_launch(void* const* d_in, const int* in_sizes, int n_in,
                              void* d_out, int out_size, void* d_ws, size_t ws_size,
                              hipStream_t stream)
{
    const float* x    = (const float*)d_in[0];
    // d_in[1] = batch indices: unused, batch = point / N by construction
    const float* W    = (const float*)d_in[2];
    const float* bias = (const float*)d_in[3];
    float* out = (float*)d_out;

    int* nbr = (int*)d_ws;   // B_*N_*K_ int32 = 2.62 MB scratch

    knn_kernel<<<B_ * (N_ / 128), 256, 0, stream>>>(x, nbr);
    edge_mlp_kernel<<<B_ * (N_ / 256), 256, 0, stream>>>(x, W, bias, nbr, out);
}